// GraphSAGE_model_83227876262250
// MI455X (gfx1250) — compile-verified
//
#include <hip/hip_runtime.h>
#include <hip/hip_bf16.h>

// ---------------------------------------------------------------------------
// GraphSAGE (3-layer, mean aggregator) for MI455X / gfx1250.
//
// Roofline: dense GEMMs ~7.8 GFLOP (negligible vs the edge scatter); the
// 25.6 MB feature table and 6.4 MB edge lists are L2-resident (192 MB L2),
// so aggregation runs at L2 bandwidth with hardware f32 atomics.  GEMMs stay
// in exact fp32 via V_WMMA_F32_16X16X4_F32 (RNE, matches the fp32 reference)
// — a bf16/fp8 path would buy nothing since compute is not the bottleneck.
//
// GEMM data movement: one block per 16-row M-tile; the block's waves cover
// all N-tiles, so H/AGG tiles (inv-deg pre-folded) and both zero-padded
// weight matrices are staged once into the WGP's 320 KB LDS.  The inner loop
// is pure ds_load + v_wmma with EXEC == all-ones (no predicated loads).
// ---------------------------------------------------------------------------

typedef __attribute__((ext_vector_type(2))) float v2f;
typedef __attribute__((ext_vector_type(8))) float v8f;

// ------------------------------- utilities ---------------------------------

__global__ void zero_f32(float* __restrict__ p, long long n) {
    long long i = (long long)blockIdx.x * blockDim.x + threadIdx.x;
    long long stride = (long long)gridDim.x * blockDim.x;
    for (; i < n; i += stride) p[i] = 0.0f;
}

__global__ void deg_count(const int* __restrict__ dst, int nE,
                          float* __restrict__ deg) {
    int i = blockIdx.x * blockDim.x + threadIdx.x;
    if (i < nE) atomicAdd(&deg[dst[i]], 1.0f);
}

__global__ void inv_deg_kernel(const float* __restrict__ deg,
                               float* __restrict__ inv, int n) {
    int i = blockIdx.x * blockDim.x + threadIdx.x;
    if (i < n) inv[i] = 1.0f / fmaxf(deg[i], 1.0f);
}

// --------------------------- edge aggregation ------------------------------
// One float4 chunk per thread; 32 consecutive threads share one edge, so the
// 512 B source row is gathered coalesced and the atomic adds are coalesced.
// All traffic is L2-resident.

__global__ void sage_scatter(const float* __restrict__ H,
                             const int* __restrict__ src,
                             const int* __restrict__ dst,
                             float* __restrict__ AGG,
                             int nE, int d) {
    const int chunks = d >> 2;                       // float4 chunks per edge
    long long total = (long long)nE * chunks;
    long long idx = (long long)blockIdx.x * blockDim.x + threadIdx.x;
    long long stride = (long long)gridDim.x * blockDim.x;
    for (; idx < total; idx += stride) {
        int e = (int)(idx / chunks);
        int c = (int)(idx % chunks);
        int s = src[e];
        int t = dst[e];
        const float4 v = *(const float4*)(H + (size_t)s * d + (size_t)c * 4);
        float* p = AGG + (size_t)t * d + (size_t)c * 4;
        atomicAdd(p + 0, v.x);
        atomicAdd(p + 1, v.y);
        atomicAdd(p + 2, v.z);
        atomicAdd(p + 3, v.w);
    }
}

// ------------------------------ WMMA GEMM ----------------------------------
// OUT[m,c] = act( sum_k H[m,k]*Ws[k,c] + (AGG[m,k]*invdeg[m])*Wn[k,c] + b[c] )
//
// Grid: blockIdx.x = M-tile (16 rows); blockDim = 32 * (DOUTP/16) waves,
// wave w computes N-tile w.  Fragment layouts (CDNA5 ISA 7.12.2, wave32):
//   A 16x4 : lane%16 = M ; VGPR0 holds K = (lane/16)*2, VGPR1 holds K+1.
//   B 4x16 : lane%16 = N ; VGPR j holds row K = j + (lane/16)*2.
//   C/D    : 8 VGPRs; VGPR r = (M = r + (lane/16)*8), N = lane%16.
// Bias is folded into the accumulator init (each lane's 8 acc elements all
// belong to one output column).

template <int DIN, int DOUTP>
__global__ void __launch_bounds__(32 * (DOUTP / 16))
sage_gemm_wmma(const float* __restrict__ H,       // [n, DIN]
               const float* __restrict__ AGG,     // [n, DIN] raw sums
               const float* __restrict__ invdeg,  // [n]
               const float* __restrict__ Ws,      // [DIN, dOut]
               const float* __restrict__ Wn,      // [DIN, dOut]
               const float* __restrict__ bias,    // [dOut]
               float* __restrict__ OUT,           // [n, dOut]
               int dOut, int relu) {
    constexpr int HSTR = DIN + 2;                 // LDS row pad: no bank conflicts
    __shared__ __align__(16) float ldsH[16 * HSTR];
    __shared__ __align__(16) float ldsG[16 * HSTR];
    __shared__ __align__(16) float ldsWs[DIN * DOUTP];
    __shared__ __align__(16) float ldsWn[DIN * DOUTP];

    const int mBase = blockIdx.x << 4;
    const int tid = threadIdx.x;
    const int nth = 32 * (DOUTP / 16);

    // Stage H tile and inv-deg-pre-scaled AGG tile (16 rows).
    for (int i = tid; i < 16 * DIN; i += nth) {
        int r = i / DIN, k = i - r * DIN;
        int row = mBase + r;
        float sc = invdeg[row];
        ldsH[r * HSTR + k] = H[(size_t)row * DIN + k];
        ldsG[r * HSTR + k] = AGG[(size_t)row * DIN + k] * sc;
    }
    // Stage weights row-major, columns zero-padded to DOUTP.
    for (int i = tid; i < DIN * DOUTP; i += nth) {
        int k = i / DOUTP, c = i - k * DOUTP;
        bool ok = (c < dOut);
        ldsWs[i] = ok ? Ws[(size_t)k * dOut + c] : 0.0f;
        ldsWn[i] = ok ? Wn[(size_t)k * dOut + c] : 0.0f;
    }
    __syncthreads();

    const int lane = tid & 31;
    const int nTile = tid >> 5;
    const int half = lane >> 4;                   // 0: lanes 0-15, 1: 16-31
    const int l16 = lane & 15;
    const int bCol = (nTile << 4) + l16;

    const float* hRow = ldsH + l16 * HSTR;
    const float* gRow = ldsG + l16 * HSTR;
    const float* wsCol = ldsWs + bCol;
    const float* wnCol = ldsWn + bCol;

    // Fold bias into accumulator (clamped load + select: no EXEC games).
    float bv = bias[bCol < dOut ? bCol : 0];
    bv = (bCol < dOut) ? bv : 0.0f;
    v8f acc = {bv, bv, bv, bv, bv, bv, bv, bv};

#pragma unroll 8
    for (int kB = 0; kB < DIN; kB += 4) {
        const int kA = kB + half * 2;             // this lane's K pair (even)
        v2f a, b;
        // self GEMM: A = H tile, B = Ws tile
        a = *(const v2f*)(hRow + kA);
        b.x = wsCol[(size_t)kA * DOUTP];
        b.y = wsCol[(size_t)(kA + 1) * DOUTP];
        acc = __builtin_amdgcn_wmma_f32_16x16x4_f32(
            /*neg_a=*/false, a, /*neg_b=*/false, b,
            /*c_mod=*/(short)0, acc, /*reuse_a=*/false, /*reuse_b=*/false);
        // neighbor GEMM: A = pre-scaled AGG tile, B = Wn tile
        a = *(const v2f*)(gRow + kA);
        b.x = wnCol[(size_t)kA * DOUTP];
        b.y = wnCol[(size_t)(kA + 1) * DOUTP];
        acc = __builtin_amdgcn_wmma_f32_16x16x4_f32(
            false, a, false, b, (short)0, acc, false, false);
    }

    if (bCol < dOut) {                            // post-WMMA predication only
#pragma unroll
        for (int r = 0; r < 8; ++r) {
            int row = mBase + r + half * 8;
            float v = acc[r];
            if (relu) v = fmaxf(v, 0.0f);
            OUT[(size_t)row * dOut + bCol] = v;
        }
    }
}

// ------------------------------- launcher ----------------------------------

static inline int cdiv(long long a, long long b) { return (int)((a + b - 1) / b); }

extern "C" void kernel_launch(void* const* d_in, const int* in_sizes, int n_in,
                              void* d_out, int out_size, void* d_ws, size_t ws_size,
                              hipStream_t stream) {
    const float* feat = (const float*)d_in[0];
    const int*   src  = (const int*)  d_in[1];
    const int*   dst  = (const int*)  d_in[2];
    const float* Ws0  = (const float*)d_in[3];
    const float* Wn0  = (const float*)d_in[4];
    const float* b0   = (const float*)d_in[5];
    const float* Ws1  = (const float*)d_in[6];
    const float* Wn1  = (const float*)d_in[7];
    const float* b1   = (const float*)d_in[8];
    const float* Ws2  = (const float*)d_in[9];
    const float* Wn2  = (const float*)d_in[10];
    const float* b2   = (const float*)d_in[11];

    const int dIn  = in_sizes[5];                 // 128
    const int dHid = in_sizes[8];                 // 128
    const int nCls = in_sizes[11];                // 47
    const int n    = in_sizes[0] / dIn;           // 50000 (multiple of 16)
    const int nE   = in_sizes[1];                 // 800000

    // workspace layout (floats):
    //   deg[n] | invdeg[n] | agg[n*dHid] | hA[n*dHid] | hB[n*dHid]
    float* ws     = (float*)d_ws;
    float* deg    = ws;
    float* invdeg = deg + n;
    float* agg    = invdeg + n;
    float* hA     = agg + (size_t)n * dHid;
    float* hB     = hA  + (size_t)n * dHid;
    float* out    = (float*)d_out;

    const int B = 256;
    const long long aggN = (long long)n * dHid;
    const int zBlocksDeg = cdiv(n, B);
    const int zBlocksAgg = cdiv(aggN, B);
    const int scBlocks   = cdiv((long long)nE * (dHid >> 2), B);

    // degree + reciprocal (once)
    zero_f32<<<zBlocksDeg, B, 0, stream>>>(deg, n);
    deg_count<<<cdiv(nE, B), B, 0, stream>>>(dst, nE, deg);
    inv_deg_kernel<<<zBlocksDeg, B, 0, stream>>>(deg, invdeg, n);

    const int nTilesM = n >> 4;                   // 3125 blocks

    // layer 0: features -> hA  (dOut=128, relu)
    zero_f32<<<zBlocksAgg, B, 0, stream>>>(agg, aggN);
    sage_scatter<<<scBlocks, B, 0, stream>>>(feat, src, dst, agg, nE, dIn);
    sage_gemm_wmma<128, 128><<<nTilesM, 256, 0, stream>>>(
        feat, agg, invdeg, Ws0, Wn0, b0, hA, dHid, 1);

    // layer 1: hA -> hB  (dOut=128, relu)
    zero_f32<<<zBlocksAgg, B, 0, stream>>>(agg, aggN);
    sage_scatter<<<scBlocks, B, 0, stream>>>(hA, src, dst, agg, nE, dHid);
    sage_gemm_wmma<128, 128><<<nTilesM, 256, 0, stream>>>(
        hA, agg, invdeg, Ws1, Wn1, b1, hB, dHid, 1);

    // layer 2: hB -> logits  (dOut=47 padded to 48, no relu)
    zero_f32<<<zBlocksAgg, B, 0, stream>>>(agg, aggN);
    sage_scatter<<<scBlocks, B, 0, stream>>>(hB, src, dst, agg, nE, dHid);
    sage_gemm_wmma<128, 48><<<nTilesM, 96, 0, stream>>>(
        hB, agg, invdeg, Ws2, Wn2, b2, out, nCls, 0);
}